// CrossAttentionTransformer_71176198029335
// MI455X (gfx1250) — compile-verified
//
#include <hip/hip_runtime.h>
#include <math.h>

// ---- problem constants (match reference) ----
constexpr int B_   = 4;
constexpr int C_   = 16;
constexpr int L_   = 1024;
constexpr int U_   = 32;    // U_SELF
constexpr int UC_  = 16;    // U_CROSS
constexpr int HID_ = 128;
// band: j in [i-63, i+64]  (128 columns = 4 x 32 lanes)

typedef __attribute__((ext_vector_type(2))) float v2f;
typedef __attribute__((ext_vector_type(8))) float v8f;

#define DEVFN __device__ __forceinline__

DEVFN float dev_tanh(float x) {
#if __has_builtin(__builtin_amdgcn_tanhf)
  return __builtin_amdgcn_tanhf(x);   // v_tanh_f32 (CDNA5 TRANS op)
#else
  return tanhf(x);
#endif
}

DEVFN float wred_max(float v) {
  for (int o = 16; o; o >>= 1) v = fmaxf(v, __shfl_xor(v, o, 32));
  return v;
}
DEVFN float wred_sum(float v) {
  for (int o = 16; o; o >>= 1) v += __shfl_xor(v, o, 32);
  return v;
}

// ---------------------------------------------------------------------------
// 1) q = xt@Wt, k = xt@Wx  for the additive self-attention.
//    x: (B,C,L) ch-major.  Q,K: (B,L,U).
// ---------------------------------------------------------------------------
__global__ void qk_kernel(const float* __restrict__ x,
                          const float* __restrict__ Wt,
                          const float* __restrict__ Wx,
                          float* __restrict__ Q, float* __restrict__ K) {
  int idx = blockIdx.x * blockDim.x + threadIdx.x;
  if (idx >= B_ * L_ * U_) return;
  int u = idx % U_;
  int l = (idx / U_) % L_;
  int b = idx / (U_ * L_);
  float q = 0.f, k = 0.f;
#pragma unroll
  for (int c = 0; c < C_; ++c) {
    float xv = x[(b * C_ + c) * L_ + l];
    q += xv * Wt[c * U_ + u];
    k += xv * Wx[c * U_ + u];
  }
  Q[idx] = q;
  K[idx] = k;
}

// ---------------------------------------------------------------------------
// 2) Banded additive self-attention. One wave32 per query i.
//    e_ij = Wa . tanh(q_i + k_j + bh) + ba  for j in [i-63, i+64]
//    a = exp(e - max)*band / (sum + 1e-5);  out[b,c,i] = sum_j a_ij x[b,c,j]
// ---------------------------------------------------------------------------
__global__ void self_attn_kernel(const float* __restrict__ x,
                                 const float* __restrict__ Q,
                                 const float* __restrict__ K,
                                 const float* __restrict__ bh,
                                 const float* __restrict__ Wa,
                                 const float* __restrict__ ba,
                                 float* __restrict__ out) {
  int wave = (blockIdx.x * blockDim.x + threadIdx.x) >> 5;
  int lane = threadIdx.x & 31;
  if (wave >= B_ * L_) return;
  int b = wave / L_;
  int i = wave % L_;

  float qv  = Q[(b * L_ + i) * U_ + lane];   // lane <-> u
  float bhv = bh[lane];
  float wav = Wa[lane];
  float bav = ba[0];

  int jlo = i - 63;
  float e[4];
  int jj[4];
  float m = -__builtin_inff();
#pragma unroll
  for (int g = 0; g < 4; ++g) {
    int j = jlo + g * 32 + lane;
    jj[g] = j;
    float ev = -__builtin_inff();
    if (j >= 0 && j < L_) {
      const float* kp = K + (size_t)(b * L_ + j) * U_;
      float acc = 0.f;
#pragma unroll
      for (int u = 0; u < U_; ++u) {
        float qq = __shfl(qv, u, 32);
        float bb = __shfl(bhv, u, 32);
        float ww = __shfl(wav, u, 32);
        acc += ww * dev_tanh(qq + kp[u] + bb);
      }
      ev = acc + bav;
    }
    e[g] = ev;
    m = fmaxf(m, ev);
  }
  m = wred_max(m);

  float p[4];
  float s = 0.f;
#pragma unroll
  for (int g = 0; g < 4; ++g) {
    int j = jj[g];
    p[g] = (j >= 0 && j < L_) ? __expf(e[g] - m) : 0.f;
    s += p[g];
  }
  s = wred_sum(s);
  float inv = 1.f / (s + 1e-5f);

#pragma unroll
  for (int c = 0; c < C_; ++c) {
    float acc = 0.f;
#pragma unroll
    for (int g = 0; g < 4; ++g) {
      int j = jj[g];
      if (j >= 0 && j < L_) acc += p[g] * x[(b * C_ + c) * L_ + j];
    }
    acc = wred_sum(acc);
    if (lane == 0) out[(b * C_ + c) * L_ + i] = acc * inv;
  }
}

// ---------------------------------------------------------------------------
// 3) out = LayerNorm_over_C(a + r) * gamma + beta   (one thread per (b,l))
// ---------------------------------------------------------------------------
__global__ void ln_add_kernel(const float* __restrict__ a,
                              const float* __restrict__ r,
                              const float* __restrict__ gamma,
                              const float* __restrict__ beta,
                              float* __restrict__ out) {
  int idx = blockIdx.x * blockDim.x + threadIdx.x;
  if (idx >= B_ * L_) return;
  int b = idx / L_;
  int l = idx % L_;
  float v[C_];
  float mean = 0.f;
#pragma unroll
  for (int c = 0; c < C_; ++c) {
    v[c] = a[(b * C_ + c) * L_ + l] + r[(b * C_ + c) * L_ + l];
    mean += v[c];
  }
  mean *= (1.f / C_);
  float var = 0.f;
#pragma unroll
  for (int c = 0; c < C_; ++c) {
    float d = v[c] - mean;
    var += d * d;
  }
  var = var * (1.f / C_) + 1e-14f;
  float inv = 1.f / sqrtf(var);
#pragma unroll
  for (int c = 0; c < C_; ++c)
    out[(b * C_ + c) * L_ + l] = (v[c] - mean) * inv * gamma[c] + beta[c];
}

// ---------------------------------------------------------------------------
// 4) Feed-forward with fp32 WMMA:  z = relu(yt@W1 + b1)@W2 + b2
//    One wave handles a 16-column tile (M=16 rows of yt).
//    FF1: 8 N-tiles x 4 K-chunks of V_WMMA_F32_16X16X4_F32 (K=16)
//    FF2: 32 K-chunks (K=128), hidden staged through LDS for D->A relayout.
// ---------------------------------------------------------------------------
__global__ void __launch_bounds__(128) ff_wmma_kernel(
    const float* __restrict__ y, const float* __restrict__ W1,
    const float* __restrict__ b1, const float* __restrict__ W2,
    const float* __restrict__ b2, float* __restrict__ z) {
  __shared__ float hs[4][16 * HID_];   // 32 KB: per-wave 16x128 hidden tile
  int w    = threadIdx.x >> 5;
  int lane = threadIdx.x & 31;
  int gcol = (blockIdx.x * 4 + w) * 16;
  int b    = gcol / L_;
  int l0   = gcol % L_;
  int mh   = lane & 15;            // M (column index) / N (output index)
  int kh   = (lane >> 4) * 2;      // K sub-offset per lane half
  int mbase = (lane >> 4) * 8;     // D-matrix row base per lane half

  // A chunks: A[m][k] = yt[l0+m][k] = y[b][k][l0+m]
  v2f a[4];
#pragma unroll
  for (int kc = 0; kc < 4; ++kc) {
    int kb = 4 * kc + kh;
    a[kc].x = y[(b * C_ + kb) * L_ + l0 + mh];
    a[kc].y = y[(b * C_ + kb + 1) * L_ + l0 + mh];
  }

  float* hw = hs[w];
#pragma unroll
  for (int t = 0; t < 8; ++t) {           // HID / 16 output tiles
    float bias = b1[t * 16 + mh];
    v8f acc;
#pragma unroll
    for (int r = 0; r < 8; ++r) acc[r] = bias;
#pragma unroll
    for (int kc = 0; kc < 4; ++kc) {
      int kb = 4 * kc + kh;
      v2f bb;
      bb.x = W1[kb * HID_ + t * 16 + mh];
      bb.y = W1[(kb + 1) * HID_ + t * 16 + mh];
      acc = __builtin_amdgcn_wmma_f32_16x16x4_f32(
          false, a[kc], false, bb, (short)0, acc, false, false);
    }
#pragma unroll
    for (int r = 0; r < 8; ++r) {
      float hv = fmaxf(acc[r], 0.f);          // relu
      hw[(mbase + r) * HID_ + t * 16 + mh] = hv;
    }
  }

  // FF2: out(16x16) = h(16x128) @ W2(128x16) + b2
  v8f acc2;
  float bias2 = b2[mh];
#pragma unroll
  for (int r = 0; r < 8; ++r) acc2[r] = bias2;
#pragma unroll
  for (int kc = 0; kc < 32; ++kc) {
    int kb = 4 * kc + kh;
    v2f aa, bb;
    aa.x = hw[mh * HID_ + kb];
    aa.y = hw[mh * HID_ + kb + 1];
    bb.x = W2[kb * C_ + mh];
    bb.y = W2[(kb + 1) * C_ + mh];
    acc2 = __builtin_amdgcn_wmma_f32_16x16x4_f32(
        false, aa, false, bb, (short)0, acc2, false, false);
  }
#pragma unroll
  for (int r = 0; r < 8; ++r)
    z[(b * C_ + mh) * L_ + l0 + mbase + r] = acc2[r];
}

// ---------------------------------------------------------------------------
// 5) Cross-attention q/k/v projections: (B,L,16) each.
// ---------------------------------------------------------------------------
__global__ void cross_qkv_kernel(const float* __restrict__ y1,
                                 const float* __restrict__ y2,
                                 const float* __restrict__ Wq,
                                 const float* __restrict__ Wk,
                                 const float* __restrict__ Wv,
                                 float* __restrict__ q, float* __restrict__ k,
                                 float* __restrict__ v) {
  int idx = blockIdx.x * blockDim.x + threadIdx.x;
  if (idx >= B_ * L_ * UC_) return;
  int u = idx % UC_;
  int l = (idx / UC_) % L_;
  int b = idx / (UC_ * L_);
  float qa = 0.f, ka = 0.f, va = 0.f;
#pragma unroll
  for (int c = 0; c < C_; ++c) {
    float a1 = y1[(b * C_ + c) * L_ + l];
    float a2 = y2[(b * C_ + c) * L_ + l];
    qa += a1 * Wq[c * UC_ + u];
    ka += a2 * Wk[c * UC_ + u];
    va += a2 * Wv[c * UC_ + u];
  }
  q[idx] = qa;
  k[idx] = ka;
  v[idx] = va;
}

// ---------------------------------------------------------------------------
// 6) Banded dot-product cross-attention. One wave per query i.
// ---------------------------------------------------------------------------
__global__ void cross_attn_kernel(const float* __restrict__ q,
                                  const float* __restrict__ k,
                                  const float* __restrict__ v,
                                  float* __restrict__ out) {
  int wave = (blockIdx.x * blockDim.x + threadIdx.x) >> 5;
  int lane = threadIdx.x & 31;
  if (wave >= B_ * L_) return;
  int b = wave / L_;
  int i = wave % L_;

  float qv = (lane < UC_) ? q[(b * L_ + i) * UC_ + lane] : 0.f;
  int jlo = i - 63;
  float e[4];
  int jj[4];
  float m = -__builtin_inff();
#pragma unroll
  for (int g = 0; g < 4; ++g) {
    int j = jlo + g * 32 + lane;
    jj[g] = j;
    float ev = -__builtin_inff();
    if (j >= 0 && j < L_) {
      const float* kp = k + (size_t)(b * L_ + j) * UC_;
      float acc = 0.f;
#pragma unroll
      for (int u = 0; u < UC_; ++u) acc += __shfl(qv, u, 32) * kp[u];
      ev = acc * 0.25f;   // 1/sqrt(U_CROSS)
    }
    e[g] = ev;
    m = fmaxf(m, ev);
  }
  m = wred_max(m);
  float p[4];
  float s = 0.f;
#pragma unroll
  for (int g = 0; g < 4; ++g) {
    int j = jj[g];
    p[g] = (j >= 0 && j < L_) ? __expf(e[g] - m) : 0.f;
    s += p[g];
  }
  s = wred_sum(s);
  float inv = 1.f / s;

#pragma unroll
  for (int c = 0; c < C_; ++c) {
    float acc = 0.f;
#pragma unroll
    for (int g = 0; g < 4; ++g) {
      int j = jj[g];
      if (j >= 0 && j < L_) acc += p[g] * v[(size_t)(b * L_ + j) * UC_ + c];
    }
    acc = wred_sum(acc);
    if (lane == 0) out[(b * C_ + c) * L_ + i] = acc * inv;
  }
}

// ---------------------------------------------------------------------------
extern "C" void kernel_launch(void* const* d_in, const int* in_sizes, int n_in,
                              void* d_out, int out_size, void* d_ws,
                              size_t ws_size, hipStream_t stream) {
  (void)in_sizes; (void)n_in; (void)out_size; (void)ws_size;

  const float* x[2]  = {(const float*)d_in[0], (const float*)d_in[1]};
  const int tb[2]    = {2, 15};   // per-stream transformer param base index
  const float* caWq  = (const float*)d_in[28];
  const float* caWk  = (const float*)d_in[29];
  const float* caWv  = (const float*)d_in[30];
  const float* tg1   = (const float*)d_in[31];
  const float* tbe1  = (const float*)d_in[32];
  const float* tW1   = (const float*)d_in[33];
  const float* tb1   = (const float*)d_in[34];
  const float* tW2   = (const float*)d_in[35];
  const float* tb2   = (const float*)d_in[36];
  const float* tg2   = (const float*)d_in[37];
  const float* tbe2  = (const float*)d_in[38];

  // workspace carve-up (floats)
  float* w = (float*)d_ws;
  size_t off = 0;
  auto alloc = [&](size_t n) { float* p = w + off; off += n; return p; };
  float* Q    = alloc((size_t)B_ * L_ * U_);
  float* K    = alloc((size_t)B_ * L_ * U_);
  float* sa   = alloc((size_t)B_ * C_ * L_);
  float* yA   = alloc((size_t)B_ * C_ * L_);
  float* zf   = alloc((size_t)B_ * C_ * L_);
  float* yB0  = alloc((size_t)B_ * C_ * L_);
  float* yB1  = alloc((size_t)B_ * C_ * L_);
  float* cq   = alloc((size_t)B_ * L_ * UC_);
  float* ck   = alloc((size_t)B_ * L_ * UC_);
  float* cv   = alloc((size_t)B_ * L_ * UC_);
  float* co   = alloc((size_t)B_ * C_ * L_);
  float* yt   = alloc((size_t)B_ * C_ * L_);
  float* yB[2] = {yB0, yB1};

  const int ln_blocks  = (B_ * L_ + 255) / 256;
  const int ff_blocks  = B_ * L_ / 64;        // 4 waves x 16 cols per block
  const int atn_blocks = B_ * L_ / 8;         // 8 waves per block

  for (int s = 0; s < 2; ++s) {
    const float* Wt  = (const float*)d_in[tb[s] + 0];
    const float* Wx  = (const float*)d_in[tb[s] + 1];
    const float* bh  = (const float*)d_in[tb[s] + 2];
    const float* Wa  = (const float*)d_in[tb[s] + 3];
    const float* ba  = (const float*)d_in[tb[s] + 4];
    const float* g1  = (const float*)d_in[tb[s] + 5];
    const float* be1 = (const float*)d_in[tb[s] + 6];
    const float* W1  = (const float*)d_in[tb[s] + 7];
    const float* b1  = (const float*)d_in[tb[s] + 8];
    const float* W2  = (const float*)d_in[tb[s] + 9];
    const float* b2  = (const float*)d_in[tb[s] + 10];
    const float* g2  = (const float*)d_in[tb[s] + 11];
    const float* be2 = (const float*)d_in[tb[s] + 12];

    qk_kernel<<<(B_ * L_ * U_ + 255) / 256, 256, 0, stream>>>(x[s], Wt, Wx, Q, K);
    self_attn_kernel<<<atn_blocks, 256, 0, stream>>>(x[s], Q, K, bh, Wa, ba, sa);
    ln_add_kernel<<<ln_blocks, 256, 0, stream>>>(sa, x[s], g1, be1, yA);
    ff_wmma_kernel<<<ff_blocks, 128, 0, stream>>>(yA, W1, b1, W2, b2, zf);
    ln_add_kernel<<<ln_blocks, 256, 0, stream>>>(zf, yA, g2, be2, yB[s]);
  }

  cross_qkv_kernel<<<(B_ * L_ * UC_ + 255) / 256, 256, 0, stream>>>(
      yB0, yB1, caWq, caWk, caWv, cq, ck, cv);
  cross_attn_kernel<<<atn_blocks, 256, 0, stream>>>(cq, ck, cv, co);
  ln_add_kernel<<<ln_blocks, 256, 0, stream>>>(co, yB0, tg1, tbe1, yt);
  ff_wmma_kernel<<<ff_blocks, 128, 0, stream>>>(yt, tW1, tb1, tW2, tb2, zf);
  ln_add_kernel<<<ln_blocks, 256, 0, stream>>>(zf, yt, tg2, tbe2, (float*)d_out);
}